// PMMs_46188078301714
// MI455X (gfx1250) — compile-verified
//
#include <hip/hip_runtime.h>
#include <math.h>

// Problem constants (fixed by the reference's setup_inputs).
constexpr int kB  = 16;    // batch
constexpr int kC  = 512;   // channels
constexpr int kN  = 4096;  // 64*64 pixels
constexpr int kK  = 3;     // NUM_PRO
constexpr int kStages = 10;
constexpr float kKappa = 20.0f;
constexpr float kEps   = 1e-6f;

// d_out layout (flat concat in return order, all f32):
//   mu_f : [16,3,512]  at 0
//   mu_b : [16,3,512]  at 24576
//   Prob : [16,2,4096] at 49152
//   P    : [16,6,4096] at 180224
constexpr int kMufOff = 0;
constexpr int kMubOff = kB * kK * kC;            // 24576
constexpr int kPmOff  = 2 * kMubOff;             // 49152
constexpr int kPOff   = kPmOff + kB * 2 * kN;    // 180224

typedef float v2f __attribute__((ext_vector_type(2)));
typedef float v8f __attribute__((ext_vector_type(8)));

__device__ __forceinline__ float wave_red_sum32(float v) {
  v += __shfl_xor(v, 1, 32);
  v += __shfl_xor(v, 2, 32);
  v += __shfl_xor(v, 4, 32);
  v += __shfl_xor(v, 8, 32);
  v += __shfl_xor(v, 16, 32);
  return v;
}

__device__ __forceinline__ float half_red_sum16(float v) {
  v += __shfl_xor(v, 1, 16);
  v += __shfl_xor(v, 2, 16);
  v += __shfl_xor(v, 4, 16);
  v += __shfl_xor(v, 8, 16);
  return v;
}

// ---------------------------------------------------------------------------
// EM kernel: one block per (batch, fg/bg). 1024 threads = 32 waves.
// Per-batch features (8 MB) are L2-resident; z/mu live in LDS.
// Prototypes stored with stride 4 and a permanently-zero 4th slot so the
// WMMA A-operand load is branch-free (lanes m16>=3 read the zero slot).
// B-operand loads are staged into an unrolled private array so the 16 loads
// per K-step issue as one clause (staggered s_wait_loadcnt, not serialized).
// ---------------------------------------------------------------------------
__global__ __launch_bounds__(1024)
void pmms_em_kernel(const float* __restrict__ sf, const float* __restrict__ mask,
                    const float* __restrict__ mu_g, float* __restrict__ out) {
  __shared__ float mu_pad[kC * 4];     // current prototypes, [c][k], k=3 slot = 0
  __shared__ float z_lds[kN * kK];     // z * mask, [n][k]
  __shared__ float acc_lds[kC * kK];   // mu accumulator / staging, [c][k]
  __shared__ float cs_part[32 * kK];   // per-wave colsum partials
  __shared__ float colsum[kK];
  __shared__ float norm_lds[kK];

  const int tid  = threadIdx.x;
  const int lane = tid & 31;
  const int wv   = tid >> 5;           // 0..31
  const int blk  = blockIdx.x;         // 0..31
  const int b    = blk >> 1;
  const bool fg  = (blk & 1) == 0;

  const float* x  = sf + (size_t)b * kC * kN;
  const float* mk = mask + (size_t)b * kN;

  __builtin_prefetch(x + (size_t)tid * 16, 0, 0);  // global_prefetch_b8

  // ---- init: stage raw mu into acc_lds, zero the pad slot once.
  for (int i = tid; i < kC * kK; i += 1024) acc_lds[i] = mu_g[i];
  for (int c = tid; c < kC; c += 1024) mu_pad[c * 4 + 3] = 0.f;
  __syncthreads();
  if (wv == 0) {
    for (int k = 0; k < kK; ++k) {
      float s = 0.f;
      for (int c = lane; c < kC; c += 32) {
        float v = acc_lds[c * kK + k];
        s += v * v;
      }
      s = wave_red_sum32(s);
      if (lane == 0) norm_lds[k] = sqrtf(s);
    }
  }
  __syncthreads();
  for (int i = tid; i < kC * kK; i += 1024) {
    int c = i / kK, k = i % kK;
    mu_pad[c * 4 + k] = acc_lds[i] / (kEps + norm_lds[k]);
  }

  // Per-tile mask values (constant across stages). Pixel index = lane&15.
  const int m16 = lane & 15;
  const int kkb = (lane >> 4) * 2;           // A/B K-offset for this half-wave
  const int arow = (m16 < 3) ? m16 : 3;      // padded prototype slot (3 => zero)
  float mkv[8];
  int pix0[8];
#pragma unroll
  for (int t = 0; t < 8; ++t) {
    pix0[t] = wv * 128 + t * 16;
    float m = mk[pix0[t] + m16];
    mkv[t] = fg ? m : (1.f - m);
  }
  __syncthreads();

  const v8f vzero = {0.f, 0.f, 0.f, 0.f, 0.f, 0.f, 0.f, 0.f};

  for (int stage = 0; stage < kStages; ++stage) {
    // ---- step 1: logits[n][k] = x_masked^T . mu via WMMA f32 16x16x4.
    // A = prototypes (rows M=0..2 of 16, rest zero via pad slot),
    // B = 16-pixel tile of masked features.
    // D layout => lanes 0..15 hold the 3 logits of their pixel in acc[0..2].
    v8f acc[8];
#pragma unroll
    for (int t = 0; t < 8; ++t) acc[t] = vzero;

    for (int c0 = 0; c0 < kC; c0 += 4) {
      const float* xr0 = x + (size_t)(c0 + kkb) * kN;
      // Stage all 16 B-operand loads first: independent destinations =>
      // one load clause with staggered waits (max memory-level parallelism).
      float bx[8], by[8];
#pragma unroll
      for (int t = 0; t < 8; ++t) {
        bx[t] = xr0[pix0[t] + m16];
        by[t] = xr0[kN + pix0[t] + m16];
      }
      v2f a;
      a.x = mu_pad[(c0 + kkb) * 4 + arow];        // branch-free (pad slot = 0)
      a.y = mu_pad[(c0 + kkb + 1) * 4 + arow];
#pragma unroll
      for (int t = 0; t < 8; ++t) {
        v2f bm;
        bm.x = bx[t] * mkv[t];
        bm.y = by[t] * mkv[t];
        acc[t] = __builtin_amdgcn_wmma_f32_16x16x4_f32(
            false, a, false, bm, (short)0, acc[t], false, false);
      }
    }

    // Softmax over k=3 per pixel (in-lane), store masked z, colsum partials.
    float cs0 = 0.f, cs1 = 0.f, cs2 = 0.f;
    if (lane < 16) {
#pragma unroll
      for (int t = 0; t < 8; ++t) {
        float s0 = kKappa * acc[t][0];
        float s1 = kKappa * acc[t][1];
        float s2 = kKappa * acc[t][2];
        float mx = fmaxf(s0, fmaxf(s1, s2));
        float e0 = __expf(s0 - mx), e1 = __expf(s1 - mx), e2 = __expf(s2 - mx);
        float inv = 1.f / (e0 + e1 + e2);
        float z0 = e0 * inv, z1 = e1 * inv, z2 = e2 * inv;
        cs0 += z0; cs1 += z1; cs2 += z2;
        int p = pix0[t] + lane;
        z_lds[p * kK + 0] = z0 * mkv[t];
        z_lds[p * kK + 1] = z1 * mkv[t];
        z_lds[p * kK + 2] = z2 * mkv[t];
      }
      cs0 = half_red_sum16(cs0);
      cs1 = half_red_sum16(cs1);
      cs2 = half_red_sum16(cs2);
      if (lane == 0) {
        cs_part[wv * kK + 0] = cs0;
        cs_part[wv * kK + 1] = cs1;
        cs_part[wv * kK + 2] = cs2;
      }
    }
    __syncthreads();
    // Deterministic colsum reduction (fixed order; no float atomics).
    if (tid < kK) {
      float s = 0.f;
      for (int w = 0; w < 32; ++w) s += cs_part[w * kK + tid];
      colsum[tid] = s;
    }
    __syncthreads();

    // ---- step 2: mu'[c][k] = sum_n sf[c][n] * (z*mask)[n][k] / (eps+colsum)
    const float i0 = 1.f / (kEps + colsum[0]);
    const float i1 = 1.f / (kEps + colsum[1]);
    const float i2 = 1.f / (kEps + colsum[2]);
    for (int cr = 0; cr < 16; ++cr) {
      const int c = wv * 16 + cr;
      const float* xr = x + (size_t)c * kN;
      float a0 = 0.f, a1 = 0.f, a2 = 0.f;
      for (int nn = lane; nn < kN; nn += 32) {
        float xv = xr[nn];  // coalesced 128B per wave iteration (L2 hit)
        a0 += xv * z_lds[nn * kK + 0];  // stride-3 dwords: conflict-free
        a1 += xv * z_lds[nn * kK + 1];
        a2 += xv * z_lds[nn * kK + 2];
      }
      a0 = wave_red_sum32(a0);
      a1 = wave_red_sum32(a1);
      a2 = wave_red_sum32(a2);
      if (lane == 0) {
        acc_lds[c * kK + 0] = a0 * i0;
        acc_lds[c * kK + 1] = a1 * i1;
        acc_lds[c * kK + 2] = a2 * i2;
      }
    }
    __syncthreads();

    // ---- L2-normalize over c.
    if (wv == 0) {
      for (int k = 0; k < kK; ++k) {
        float s = 0.f;
        for (int c = lane; c < kC; c += 32) {
          float v = acc_lds[c * kK + k];
          s += v * v;
        }
        s = wave_red_sum32(s);
        if (lane == 0) norm_lds[k] = sqrtf(s);
      }
    }
    __syncthreads();
    for (int i = tid; i < kC * kK; i += 1024) {
      int c = i / kK, k = i % kK;
      mu_pad[c * 4 + k] = acc_lds[i] / (kEps + norm_lds[k]);
    }
    __syncthreads();
  }

  // Write mu^T (output layout [k][c]).
  float* mo = out + (fg ? kMufOff : kMubOff) + (size_t)b * kK * kC;
  for (int i = tid; i < kC * kK; i += 1024) {
    int k = i / kC, c = i % kC;
    mo[i] = mu_pad[c * 4 + k];
  }
}

// ---------------------------------------------------------------------------
// Query readout: logits[n][6] = xq^T . [mu_f; mu_b], softmax over 6.
// 256 blocks = 16 batches x 16 pixel segments, 256 threads (8 waves x 2 tiles).
// mu_cat padded to 8 rows (rows 6..7 zero) => branch-free A loads.
// ---------------------------------------------------------------------------
__global__ __launch_bounds__(256)
void pmms_query_kernel(const float* __restrict__ qf, const float* __restrict__ muf,
                       const float* __restrict__ mub, float* __restrict__ out) {
  __shared__ float mc[8 * kC];  // [r][c]; r = 0..2 fg, 3..5 bg, 6..7 zero pad

  const int tid  = threadIdx.x;
  const int lane = tid & 31;
  const int wv   = tid >> 5;      // 0..7
  const int b    = blockIdx.x >> 4;
  const int seg  = blockIdx.x & 15;

  const float* xq = qf + (size_t)b * kC * kN;
  const float* mf = muf + (size_t)b * kK * kC;  // [k][c]
  const float* mb = mub + (size_t)b * kK * kC;
  for (int i = tid; i < kK * kC; i += 256) {
    mc[i] = mf[i];
    mc[kK * kC + i] = mb[i];
  }
  for (int i = tid; i < 2 * kC; i += 256) mc[6 * kC + i] = 0.f;  // pad rows
  __syncthreads();

  const int m16 = lane & 15;
  const int kkb = (lane >> 4) * 2;
  const int arow = (m16 < 6) ? m16 : 6;   // padded row (6 => zero)
  int pix0[2];
#pragma unroll
  for (int t = 0; t < 2; ++t) pix0[t] = seg * 256 + wv * 32 + t * 16;

  const v8f vzero = {0.f, 0.f, 0.f, 0.f, 0.f, 0.f, 0.f, 0.f};
  v8f acc[2];
  acc[0] = vzero;
  acc[1] = vzero;

  for (int c0 = 0; c0 < kC; c0 += 4) {
    const float* xr0 = xq + (size_t)(c0 + kkb) * kN;
    float bx[2], by[2];
#pragma unroll
    for (int t = 0; t < 2; ++t) {
      bx[t] = xr0[pix0[t] + m16];
      by[t] = xr0[kN + pix0[t] + m16];
    }
    v2f a;
    a.x = mc[arow * kC + c0 + kkb];       // branch-free (pad row = 0)
    a.y = mc[arow * kC + c0 + kkb + 1];
#pragma unroll
    for (int t = 0; t < 2; ++t) {
      v2f bm;
      bm.x = bx[t];
      bm.y = by[t];
      acc[t] = __builtin_amdgcn_wmma_f32_16x16x4_f32(
          false, a, false, bm, (short)0, acc[t], false, false);
    }
  }

  if (lane < 16) {
    float* P  = out + kPOff + (size_t)b * 6 * kN;
    float* PM = out + kPmOff + (size_t)b * 2 * kN;
#pragma unroll
    for (int t = 0; t < 2; ++t) {
      const int p = pix0[t] + lane;
      float s[6];
      float mx = -3.4e38f;
#pragma unroll
      for (int r = 0; r < 6; ++r) {
        s[r] = acc[t][r];
        mx = fmaxf(mx, s[r]);
      }
      float sum = 0.f;
#pragma unroll
      for (int r = 0; r < 6; ++r) {
        s[r] = __expf(s[r] - mx);
        sum += s[r];
      }
      const float inv = 1.f / sum;
      float pf = 0.f, pb = 0.f;
#pragma unroll
      for (int r = 0; r < 6; ++r) {
        float z = s[r] * inv;
        P[(size_t)r * kN + p] = z;
        if (r < kK) pf += z; else pb += z;
      }
      PM[p] = pb;        // Prob_map channel 0 = background
      PM[kN + p] = pf;   // Prob_map channel 1 = foreground
    }
  }
}

extern "C" void kernel_launch(void* const* d_in, const int* in_sizes, int n_in,
                              void* d_out, int out_size, void* d_ws, size_t ws_size,
                              hipStream_t stream) {
  const float* sf   = (const float*)d_in[0];  // [16,512,64,64]
  const float* mask = (const float*)d_in[1];  // [16,1,64,64]
  const float* qf   = (const float*)d_in[2];  // [16,512,64,64]
  const float* mu   = (const float*)d_in[3];  // [1,512,3]
  float* out = (float*)d_out;

  // EM for fg+bg: 32 blocks (batch x {fg,bg}), 32 waves each.
  pmms_em_kernel<<<32, 1024, 0, stream>>>(sf, mask, mu, out);
  // Query readout reads mu_f/mu_b already written into d_out (same stream).
  pmms_query_kernel<<<256, 256, 0, stream>>>(qf, out + kMufOff, out + kMubOff, out);
}